// BoxListHNMS_49658411876613
// MI455X (gfx1250) — compile-verified
//
#include <hip/hip_runtime.h>
#include <hip/hip_bf16.h>

typedef unsigned long long u64;
typedef unsigned int u32;

#define PROBE_CAP 4096
#define COLLECT_CAP 1024

// ---- CDNA5 async global->LDS path (guarded; falls back to plain loads) ----
#if defined(__has_builtin)
#if __has_builtin(__builtin_amdgcn_global_load_async_to_lds_b64) && \
    __has_builtin(__builtin_amdgcn_s_wait_asynccnt)
#define USE_ASYNC_LDS 1
#endif
#endif

#ifdef USE_ASYNC_LDS
typedef int v2i_t __attribute__((__vector_size__(2 * sizeof(int))));
typedef __attribute__((address_space(1))) v2i_t* gasync_t;
typedef __attribute__((address_space(3))) v2i_t* lasync_t;
#endif

struct SelState {
  u64 prefix;    // accumulating bits of the k-th largest key (MSB-first)
  int shift;     // current digit shift (56 -> 0)
  u32 kremain;   // rank remaining within current prefix class
  u32 counter;   // collect counter
  u32 pad[10];
};

static __device__ __forceinline__ u64 mix64(u64 z) {
  z += 0x9E3779B97F4A7C15ull;
  z = (z ^ (z >> 30)) * 0xBF58476D1CE4E5B9ull;
  z = (z ^ (z >> 27)) * 0x94D049BB133111EBull;
  return z ^ (z >> 31);
}

// Monotonic float->uint mapping (larger score -> larger key)
static __device__ __forceinline__ u32 score_key(float s) {
  u32 u = __float_as_uint(s);
  return (u & 0x80000000u) ? ~u : (u | 0x80000000u);
}

// Mirrors reference float32 math exactly:
//   qw = floor(log(w)/log(a) + off); dw = a^qw; qx = floor(cx/(step*dw)+off)
//   key_hi = (qw+64)*128+(qh+64); key_lo = qx*65536+qy
static __device__ __forceinline__ u64 cell_key(float4 r, int m, int num) {
  const float alpha = 0.71f;
  const float log_a = logf(0.71f);                // constant-folded float32
  const float step  = (float)(1.0 / 0.71 - 1.0);  // python double -> f32 use
  float off = (float)((double)m / (double)num);
  int qw = (int)floorf(logf(r.z) / log_a + off);
  int qh = (int)floorf(logf(r.w) / log_a + off);
  float dw = powf(alpha, (float)qw);
  float dh = powf(alpha, (float)qh);
  int qx = (int)floorf(r.x / (step * dw) + off);
  int qy = (int)floorf(r.y / (step * dh) + off);
  u32 key_hi = (u32)((qw + 64) * 128 + (qh + 64));
  u32 key_lo = (u32)(qx * 65536 + qy);
  // key_hi is small (boxes bounded) -> pack (key_hi | m | key_lo) into 64b
  return (((u64)key_hi) << 38) | (((u64)(m & 63)) << 32) | (u64)key_lo;
}

__global__ void init_table_kernel(u64* __restrict__ hkey, u64* __restrict__ hval, u32 tab) {
  for (u32 i = blockIdx.x * blockDim.x + threadIdx.x; i < tab; i += gridDim.x * blockDim.x) {
    hkey[i] = ~0ull;
    hval[i] = 0ull;
  }
}

__global__ void init_state_kernel(SelState* st, u32* __restrict__ hist, const int* __restrict__ pk) {
  if (threadIdx.x == 0) {
    int k = *pk;
    if (k < 1) k = 1;
    if (k > COLLECT_CAP) k = COLLECT_CAP;
    st->prefix = 0ull;
    st->shift = 56;
    st->kremain = (u32)k;
    st->counter = 0u;
  }
  for (int i = threadIdx.x; i < 256; i += blockDim.x) hist[i] = 0u;
}

// Insert: claim cell slot via open addressing, then atomicMax packed winner.
__global__ void hnms_insert_kernel(const float4* __restrict__ rects,
                                   const float* __restrict__ scores, int n,
                                   const int* __restrict__ pnum,
                                   u64* __restrict__ hkey, u64* __restrict__ hval,
                                   u32 tabmask) {
  int num = *pnum;
  for (int i = blockIdx.x * blockDim.x + threadIdx.x; i < n; i += gridDim.x * blockDim.x) {
    float4 r = rects[i];
    u64 val = (((u64)score_key(scores[i])) << 32) | (u64)(u32)(~(u32)i);
    for (int m = 0; m < num; ++m) {
      u64 key = cell_key(r, m, num);
      u32 p = (u32)mix64(key) & tabmask;
      int budget = PROBE_CAP;
      for (;;) {
        u64 k = hkey[p];
        if (k == ~0ull) k = atomicCAS(&hkey[p], ~0ull, key);  // k = old value
        if (k == ~0ull || k == key) break;                    // claimed or matched
        if (--budget == 0) break;                             // safety fallback
        p = (p + 1) & tabmask;
      }
      atomicMax(&hval[p], val);
    }
  }
}

// Check: survive iff winner in all `num` cells; emit 64-bit sort key (0 = dead).
__global__ void hnms_check_kernel(const float4* __restrict__ rects,
                                  const float* __restrict__ scores, int n,
                                  const int* __restrict__ pnum,
                                  const u64* __restrict__ hkey, const u64* __restrict__ hval,
                                  u32 tabmask, u64* __restrict__ skeys) {
  int num = *pnum;
  for (int i = blockIdx.x * blockDim.x + threadIdx.x; i < n; i += gridDim.x * blockDim.x) {
    float4 r = rects[i];
    u64 val = (((u64)score_key(scores[i])) << 32) | (u64)(u32)(~(u32)i);
    bool keep = true;
    for (int m = 0; m < num && keep; ++m) {
      u64 key = cell_key(r, m, num);
      u32 p = (u32)mix64(key) & tabmask;
      int budget = PROBE_CAP;
      for (;;) {
        u64 k = hkey[p];
        if (k == key || k == ~0ull) break;
        if (--budget == 0) break;
        p = (p + 1) & tabmask;
      }
      keep = (hval[p] == val);
    }
    skeys[i] = keep ? val : 0ull;
  }
}

// One radix-select digit pass: 256-bin LDS histogram of elements matching prefix.
__global__ void hist_kernel(const u64* __restrict__ keys, int n,
                            const SelState* __restrict__ st, u32* __restrict__ hist) {
  __shared__ u32 lh[256];
  for (int i = threadIdx.x; i < 256; i += blockDim.x) lh[i] = 0u;
  __syncthreads();
  int shift = st->shift;
  u64 prefix = st->prefix;
  u64 hm = (shift >= 56) ? 0ull : (~0ull << (shift + 8));
  int stride = gridDim.x * blockDim.x;
  for (int i = blockIdx.x * blockDim.x + threadIdx.x; i < n; i += stride) {
    __builtin_prefetch(keys + i + stride, 0, 1);  // global_prefetch_b8
    u64 k = keys[i];
    if ((k & hm) == prefix) atomicAdd(&lh[(u32)(k >> shift) & 255u], 1u);
  }
  __syncthreads();
  for (int i = threadIdx.x; i < 256; i += blockDim.x)
    if (lh[i]) atomicAdd(&hist[i], lh[i]);
}

// Pick the digit containing the k-th largest; update state; clear histogram.
__global__ void scan_kernel(SelState* st, u32* __restrict__ hist) {
  if (threadIdx.x == 0) {
    int shift = st->shift;
    u32 kr = st->kremain;
    u32 acc = 0;
    int b = 255;
    for (; b > 0; --b) {
      u32 c = hist[b];
      if (acc + c >= kr) break;
      acc += c;
    }
    st->kremain = kr - acc;
    st->prefix |= ((u64)(u32)b) << shift;
    st->shift = shift - 8;
  }
  __syncthreads();
  for (int i = threadIdx.x; i < 256; i += blockDim.x) hist[i] = 0u;
}

// Collect everything >= threshold T (keys are unique, so exactly k when T>0).
__global__ void collect_kernel(const u64* __restrict__ keys, int n,
                               SelState* st, u64* __restrict__ buf) {
  u64 T = st->prefix;
  for (int i = blockIdx.x * blockDim.x + threadIdx.x; i < n; i += gridDim.x * blockDim.x) {
    u64 k = keys[i];
    if (k != 0ull && k >= T) {
      u32 p = atomicAdd(&st->counter, 1u);
      if (p < COLLECT_CAP) buf[p] = k;
    }
  }
}

// Single 1024-thread block: stage winners into LDS (async if available),
// bitonic-sort descending, emit [rows,5] output (zero rows for empty slots).
__global__ void __launch_bounds__(1024)
sortout_kernel(const SelState* __restrict__ st, const u64* __restrict__ buf,
               const float4* __restrict__ rects, const float* __restrict__ scores,
               float* __restrict__ out, int rows) {
  __shared__ u64 sk[COLLECT_CAP];
  int t = threadIdx.x;
  u32 cnt = st->counter;
  if (cnt > COLLECT_CAP) cnt = COLLECT_CAP;
#ifdef USE_ASYNC_LDS
  {
    // CDNA5 async global->LDS copy (ASYNCcnt path), one b64 per lane.
    __builtin_amdgcn_global_load_async_to_lds_b64(
        (gasync_t)(u64*)(buf + t), (lasync_t)(sk + t), 0, 0);
    __builtin_amdgcn_s_wait_asynccnt(0);
  }
  __syncthreads();
  if (t >= (int)cnt) sk[t] = 0ull;  // mask off unused/garbage entries
  __syncthreads();
#else
  sk[t] = (t < (int)cnt) ? buf[t] : 0ull;
  __syncthreads();
#endif
  // Bitonic sort, descending (1024 elements, 1024 threads)
  for (int k = 2; k <= COLLECT_CAP; k <<= 1) {
    for (int j = k >> 1; j > 0; j >>= 1) {
      int ixj = t ^ j;
      if (ixj > t) {
        u64 a = sk[t], b = sk[ixj];
        bool desc = ((t & k) == 0);
        if (desc ? (a < b) : (a > b)) { sk[t] = b; sk[ixj] = a; }
      }
      __syncthreads();
    }
  }
  if (t < rows) {
    u64 key = sk[t];
    float o0 = 0.f, o1 = 0.f, o2 = 0.f, o3 = 0.f, o4 = 0.f;
    if (key != 0ull) {
      u32 idx = ~((u32)(key & 0xFFFFFFFFull));
      float4 r = rects[idx];
      o0 = r.x; o1 = r.y; o2 = r.z; o3 = r.w; o4 = scores[idx];
    }
    float* o = out + (size_t)t * 5;
    o[0] = o0; o[1] = o1; o[2] = o2; o[3] = o3; o[4] = o4;
  }
}

extern "C" void kernel_launch(void* const* d_in, const int* in_sizes, int n_in,
                              void* d_out, int out_size, void* d_ws, size_t ws_size,
                              hipStream_t stream) {
  const float4* rects = (const float4*)d_in[0];
  const float* scores = (const float*)d_in[1];
  const int* pnum = (const int*)d_in[2];
  const int* pk = (const int*)d_in[3];
  int n = in_sizes[1];
  int rows = out_size / 5;
  if (n <= 0 || rows <= 0) return;

  // Workspace carve. Hash table sized to fit; 8M slots (128MB) sits in the
  // 192MB L2 so the atomic insert phase never goes to HBM.
  size_t fixed = (size_t)n * 8 + (size_t)COLLECT_CAP * 8 + 256 * 4 + sizeof(SelState) + 256;
  size_t tab = (size_t)1 << 23;
  while (tab > ((size_t)1 << 20) && tab * 16 + fixed > ws_size) tab >>= 1;
  char* w = (char*)d_ws;
  u64* hkey = (u64*)w;      w += tab * 8;
  u64* hval = (u64*)w;      w += tab * 8;
  u64* skeys = (u64*)w;     w += (size_t)n * 8;
  u64* buf = (u64*)w;       w += (size_t)COLLECT_CAP * 8;
  u32* hist = (u32*)w;      w += 256 * 4;
  SelState* st = (SelState*)w;
  u32 tabmask = (u32)(tab - 1);

  const int tb = 256;  // 8 wave32 waves per workgroup
  int nblocks = (n + tb - 1) / tb;
  int tblocks = (int)((tab + (size_t)tb - 1) / (size_t)tb);

  hipLaunchKernelGGL(init_table_kernel, dim3(tblocks), dim3(tb), 0, stream, hkey, hval, (u32)tab);
  hipLaunchKernelGGL(init_state_kernel, dim3(1), dim3(256), 0, stream, st, hist, pk);
  hipLaunchKernelGGL(hnms_insert_kernel, dim3(nblocks), dim3(tb), 0, stream,
                     rects, scores, n, pnum, hkey, hval, tabmask);
  hipLaunchKernelGGL(hnms_check_kernel, dim3(nblocks), dim3(tb), 0, stream,
                     rects, scores, n, pnum, hkey, hval, tabmask, skeys);
  for (int p = 0; p < 8; ++p) {
    hipLaunchKernelGGL(hist_kernel, dim3(2048), dim3(tb), 0, stream, skeys, n, st, hist);
    hipLaunchKernelGGL(scan_kernel, dim3(1), dim3(256), 0, stream, st, hist);
  }
  hipLaunchKernelGGL(collect_kernel, dim3(nblocks), dim3(tb), 0, stream, skeys, n, st, buf);
  hipLaunchKernelGGL(sortout_kernel, dim3(1), dim3(1024), 0, stream,
                     st, buf, rects, scores, (float*)d_out, rows);
}